// AdaptiveConv2d_38448547234213
// MI455X (gfx1250) — compile-verified
//
#include <hip/hip_runtime.h>

typedef __attribute__((ext_vector_type(2))) float v2f;
typedef __attribute__((ext_vector_type(8))) float v8f;
typedef __attribute__((ext_vector_type(4))) int   v4i;

typedef __attribute__((address_space(3))) float lds_f;
typedef __attribute__((address_space(1))) float glb_f;
typedef __attribute__((address_space(1))) int   g_i32;
typedef __attribute__((address_space(3))) int   l_i32;
typedef __attribute__((address_space(1))) v4i   g_v4i;
typedef __attribute__((address_space(3))) v4i   l_v4i;

#if __has_builtin(__builtin_amdgcn_global_load_async_to_lds_b32)
#define HAVE_ASYNC_LD 1
#else
#define HAVE_ASYNC_LD 0
#endif
#if __has_builtin(__builtin_amdgcn_global_store_async_from_lds_b128)
#define HAVE_ASYNC_ST 1
#else
#define HAVE_ASYNC_ST 0
#endif

__device__ __forceinline__ void wait_async0() {
#if __has_builtin(__builtin_amdgcn_s_wait_asynccnt)
  __builtin_amdgcn_s_wait_asynccnt(0);
#elif defined(__AMDGCN__)
  asm volatile("s_wait_asynccnt 0" ::: "memory");
#endif
}

// Problem shape from setup_inputs(): x (B,C,H,W), weight (9,O,C), bias (O)
constexpr int B = 8, C = 64, O = 64, H = 80, W = 800;
constexpr int WT    = 16;    // one M-tile (w) per WMMA
constexpr int MT    = 2;     // M-tiles per workgroup -> 32 w per WG
constexpr int WTILE = WT * MT;
constexpr int APAD  = 68;    // c-stride in LDS A buffer (64 + 4 pad -> conflict-free frag loads)
constexpr int OPAD  = 36;    // w-stride in LDS output staging (32 + 4 pad, keeps 16B align)
constexpr int WSZ   = 9 * O * C;  // weight element count

// One-time weight shuffle into fragment-ready order:
//   wt[((tap*32 + cp)*64 + o)*2 + e] = weight[(tap*64 + o)*64 + 2*cp + e]
// so the main-loop B-fragment load is a single coalesced global_load_b64.
__global__ void transpose_weight(const float* __restrict__ w, float* __restrict__ wt) {
  int i = blockIdx.x * 256 + threadIdx.x;
  if (i >= WSZ) return;
  int e   = i & 1;
  int o   = (i >> 1) & 63;
  int cp  = (i >> 7) & 31;
  int tap = i >> 12;
  wt[i] = w[(((tap << 6) | o) << 6) | ((cp << 1) | e)];
}

template <bool TW>
__launch_bounds__(128)
__global__ void adaptive_conv2d_wmma(const float* __restrict__ x,
                                     const float* __restrict__ dh,
                                     const float* __restrict__ dw,
                                     const float* __restrict__ weight,  // original [9][O][C]
                                     const float* __restrict__ wB,      // transposed (if TW)
                                     const float* __restrict__ bias,
                                     float* __restrict__ out) {
  __shared__ float lds_a[9 * WTILE * APAD];   // gathered A: [tap][wi 0..31][c]
  __shared__ float lds_o[O * OPAD];           // output staging: [o][w 0..31]

  const int w0  = blockIdx.x * WTILE;
  const int h   = blockIdx.y;
  const int b   = blockIdx.z;
  const int tid = threadIdx.x;

  // ---------------- gather phase: 9 taps x 32 w x 64 c into LDS ----------------
  for (int p = tid; p < 9 * WTILE; p += 128) {
    const int tap = p >> 5;
    const int wi  = p & 31;
    const int w   = w0 + wi;
    int dhv = (int)dh[b * W + w]; if (dhv < 1) dhv = 1;
    int dwv = (int)dw[b * W + w]; if (dwv < 1) dwv = 1;
    const int ri = tap / 3, ci = tap % 3;
    int row, col;
    if (ri == 0)      { int t = h - dhv; row = (t < 0 ? -t : t) % H; }
    else if (ri == 1) { row = h; }
    else              { int u = h + dhv; row = (u >= H) ? h : u; }
    if (ci == 0)      { int t = w - dwv; col = (t < 0 ? -t : t) % W; }
    else if (ci == 1) { col = w; }
    else              { int rt = w + dwv; col = (rt >= W) ? (W - 1 - (rt % W)) : rt; }

    const float* src = x + ((size_t)b * C * H + (size_t)row) * W + col;  // +c*H*W per channel
    float* dst = &lds_a[(tap * WTILE + wi) * APAD];
#if HAVE_ASYNC_LD && defined(__AMDGCN__)
    #pragma unroll 8
    for (int c = 0; c < C; ++c)
      __builtin_amdgcn_global_load_async_to_lds_b32(
          (g_i32*)(glb_f*)(src + (size_t)c * H * W), (l_i32*)(lds_f*)(dst + c), 0, 0);
#else
    #pragma unroll 8
    for (int c = 0; c < C; ++c) dst[c] = src[(size_t)c * H * W];
#endif
  }
#if HAVE_ASYNC_LD && defined(__AMDGCN__)
  wait_async0();
#endif
  __syncthreads();

  // ---------------- WMMA compute ----------------
  // Wave grid 2x2: mt = wave>>1 picks the 16-w M-tile, wave&1 picks an o-pair.
  const int wave  = tid >> 5;
  const int lane  = tid & 31;
  const int mt    = wave >> 1;
  const int np    = wave & 1;          // N-pair: o-tiles {2*np, 2*np+1}
  const int nloc  = lane & 15;
  const int khalf = (lane >> 4) << 1;  // 0 or 2
  const int kh2   = khalf >> 1;        // 0 or 1 (c-pair offset)
  const int o0    = np * 32 + nloc;    // first o-tile's o for this lane
  const int o1    = o0 + 16;           // second o-tile's o

  v8f acc0, acc1;
  const float bv0 = bias[o0];
  const float bv1 = bias[o1];
  #pragma unroll
  for (int r = 0; r < 8; ++r) { acc0[r] = bv0; acc1[r] = bv1; }  // fold bias into C

  for (int tap = 0; tap < 9; ++tap) {
    const float* ar = &lds_a[(tap * WTILE + mt * WT + nloc) * APAD + khalf];
    const float* wb0;
    const float* wt0;
    if (TW) {
      // coalesced: ((tap*32 + 2j + kh2)*64 + o)*2
      wb0 = wB + (size_t)(((tap * 32 + kh2) * 64 + o0) * 2);
    } else {
      wt0 = weight + ((size_t)tap * O) * C;
    }
    #pragma unroll
    for (int j = 0; j < 16; ++j) {
      v2f a = *(const v2f*)(ar + 4 * j);
      v2f b0, b1;
      if (TW) {
        b0 = *(const v2f*)(wb0 + 256 * j);        // o-tile 2*np
        b1 = *(const v2f*)(wb0 + 256 * j + 32);   // o-tile 2*np+1 (+16 o -> +32 floats)
      } else {
        b0 = *(const v2f*)(wt0 + (size_t)o0 * C + 4 * j + khalf);
        b1 = *(const v2f*)(wt0 + (size_t)o1 * C + 4 * j + khalf);
      }
      acc0 = __builtin_amdgcn_wmma_f32_16x16x4_f32(false, a, false, b0, (short)0, acc0, false, false);
      acc1 = __builtin_amdgcn_wmma_f32_16x16x4_f32(false, a, false, b1, (short)0, acc1, false, false);
    }
  }

  // ---------------- epilogue: transpose through LDS, coalesced b128 stores -----
  const int mbase = mt * WT + (lane >> 4) * 8;   // w offset within WG tile
  #pragma unroll
  for (int r = 0; r < 8; ++r) {
    lds_o[o0 * OPAD + mbase + r] = acc0[r];
    lds_o[o1 * OPAD + mbase + r] = acc1[r];
  }
  __syncthreads();

  const int oo = tid >> 1;             // 2 threads per o-row
  const int ws = (tid & 1) * 16;       // each stores 16 contiguous w
  float* op       = out + (((size_t)b * O + oo) * H + h) * W + w0 + ws;
  const float* sp = &lds_o[oo * OPAD + ws];
#if HAVE_ASYNC_ST && defined(__AMDGCN__)
  #pragma unroll
  for (int q = 0; q < 4; ++q)
    __builtin_amdgcn_global_store_async_from_lds_b128(
        (g_v4i*)(glb_f*)(op + 4 * q), (l_v4i*)(lds_f*)(sp + 4 * q), 0, 0);
  // s_endpgm performs an implicit wait-idle; async stores complete before wave exit.
#else
  #pragma unroll
  for (int q = 0; q < 4; ++q)
    *((float4*)op + q) = *((const float4*)sp + q);
#endif
}

extern "C" void kernel_launch(void* const* d_in, const int* in_sizes, int n_in,
                              void* d_out, int out_size, void* d_ws, size_t ws_size,
                              hipStream_t stream) {
  const float* x      = (const float*)d_in[0];
  const float* dh     = (const float*)d_in[1];
  const float* dw     = (const float*)d_in[2];
  const float* weight = (const float*)d_in[3];
  const float* bias   = (const float*)d_in[4];
  float* out          = (float*)d_out;

  dim3 grid(W / WTILE, H, B);   // 25 x 80 x 8 = 16000 workgroups

  if (ws_size >= (size_t)WSZ * sizeof(float)) {
    float* wB = (float*)d_ws;
    transpose_weight<<<(WSZ + 255) / 256, 256, 0, stream>>>(weight, wB);
    adaptive_conv2d_wmma<true><<<grid, 128, 0, stream>>>(x, dh, dw, weight, wB, bias, out);
  } else {
    adaptive_conv2d_wmma<false><<<grid, 128, 0, stream>>>(x, dh, dw, weight, nullptr, bias, out);
  }
}